// HiddenStateRollingBuffer_80376017977971
// MI455X (gfx1250) — compile-verified
//
#include <hip/hip_runtime.h>
#include <cstdint>

// Problem geometry (from the reference).
#define BUF_ROWS   129          // MAX_BATCH + 1
#define BUFFER_LEN 64
#define HIDDEN     4096
#define MAX_BATCH  128

#define BUF_ELEMS  ((size_t)BUF_ROWS * BUFFER_LEN * HIDDEN)   // 33,816,576 floats
#define BUF_VEC4   (BUF_ELEMS / 4)                            // 8,454,144 float4

#define COPY_THREADS 256
#define VPT          4                       // float4 transfers per thread
#define VEC4_PER_BLK (COPY_THREADS * VPT)    // 1024 float4 = 16 KB per block
#define COPY_BLOCKS  (BUF_VEC4 / VEC4_PER_BLK)   // 8256, exact

// ---------------------------------------------------------------------------
// Kernel 1: bulk copy of `buffer` -> out_buffer, streamed through LDS with the
// gfx1250 async global<->LDS data movers (ASYNCcnt-tracked, no VGPR round
// trip). Each thread owns VPT private 16B LDS slots, so no barriers are
// needed — only s_wait_asynccnt between the load batch and the store batch.
// ---------------------------------------------------------------------------
__global__ __launch_bounds__(COPY_THREADS)
void rollbuf_copy_async(const float4* __restrict__ src, float4* __restrict__ dst) {
    __shared__ alignas(16) float4 tile[VEC4_PER_BLK];

    const uint32_t t    = threadIdx.x;
    const size_t   base = (size_t)blockIdx.x * VEC4_PER_BLK;

    uint32_t lds[VPT];
    uint64_t gsrc[VPT], gdst[VPT];
#pragma unroll
    for (int j = 0; j < VPT; ++j) {
        const uint32_t slot = (uint32_t)j * COPY_THREADS + t;   // coalesced per j
        // Low 32 bits of a generic pointer to __shared__ == LDS byte offset.
        lds[j]  = (uint32_t)(uintptr_t)(&tile[slot]);
        gsrc[j] = (uint64_t)(uintptr_t)(src + base + slot);
        gdst[j] = (uint64_t)(uintptr_t)(dst + base + slot);
    }

    // Issue 4 async 16B loads (global -> LDS), per lane.
#pragma unroll
    for (int j = 0; j < VPT; ++j)
        asm volatile("global_load_async_to_lds_b128 %0, %1, off"
                     :: "v"(lds[j]), "v"(gsrc[j]) : "memory");

    // Wait for LDS to be filled by this wave's async loads.
    asm volatile("s_wait_asynccnt 0" ::: "memory");

    // Issue 4 async 16B stores (LDS -> global), per lane.
#pragma unroll
    for (int j = 0; j < VPT; ++j)
        asm volatile("global_store_async_from_lds_b128 %0, %1, off"
                     :: "v"(gdst[j]), "v"(lds[j]) : "memory");

    // Drain before the wave retires (S_ENDPGM also waits, this is belt+braces).
    asm volatile("s_wait_asynccnt 0" ::: "memory");
}

// ---------------------------------------------------------------------------
// Kernel 2: per-batch scatter of hidden_state into its (row, pos) slot of the
// output buffer, and into `gathered` (which equals hidden_state since seq_ids
// are unique). One block per batch element; 4096 floats = 1024 float4.
// Runs after kernel 1 by stream ordering, so it overwrites the copied row.
// ---------------------------------------------------------------------------
__global__ __launch_bounds__(256)
void rollbuf_scatter(const float* __restrict__ hidden,
                     const int*  __restrict__ seq_ids,
                     const int*  __restrict__ pos_ids,
                     float* __restrict__ out_buf,
                     float* __restrict__ out_gath) {
    const int b   = blockIdx.x;
    const int row = seq_ids[b];
    const int pos = (int)((unsigned)pos_ids[b] % BUFFER_LEN);

    const float4* __restrict__ src = (const float4*)(hidden + (size_t)b * HIDDEN);
    float4* __restrict__ dbuf  =
        (float4*)(out_buf + (((size_t)row * BUFFER_LEN) + (size_t)pos) * HIDDEN);
    float4* __restrict__ dgath = (float4*)(out_gath + (size_t)b * HIDDEN);

#pragma unroll
    for (int j = 0; j < 4; ++j) {
        const int i = j * 256 + threadIdx.x;     // 1024 float4 total
        const float4 v = src[i];
        dbuf[i]  = v;
        dgath[i] = v;
    }
}

extern "C" void kernel_launch(void* const* d_in, const int* in_sizes, int n_in,
                              void* d_out, int out_size, void* d_ws, size_t ws_size,
                              hipStream_t stream) {
    (void)in_sizes; (void)n_in; (void)out_size; (void)d_ws; (void)ws_size;

    const float* buffer  = (const float*)d_in[0];   // (129, 64, 4096) f32
    const float* hidden  = (const float*)d_in[1];   // (128, 1, 4096) f32
    const int*   seq_ids = (const int*)d_in[2];     // (128,) i32
    const int*   pos_ids = (const int*)d_in[3];     // (128,) i32

    float* out_buf  = (float*)d_out;                // new_buffer, flat
    float* out_gath = out_buf + BUF_ELEMS;          // gathered, flat

    // 1) Bulk copy buffer -> new_buffer (HBM-bound, ~270 MB of traffic).
    rollbuf_copy_async<<<dim3((unsigned)COPY_BLOCKS), dim3(COPY_THREADS), 0, stream>>>(
        (const float4*)buffer, (float4*)out_buf);

    // 2) Scatter hidden_state rows into their slots + emit gathered.
    rollbuf_scatter<<<dim3(MAX_BATCH), dim3(256), 0, stream>>>(
        hidden, seq_ids, pos_ids, out_buf, out_gath);
}